// CRNLayer_35510789603730
// MI455X (gfx1250) — compile-verified
//
#include <hip/hip_runtime.h>
#include <math.h>

// ---------------- problem constants (from reference) ----------------
#define N_IMG 32
#define C_IN  512
#define WDIM  32
#define HDIM  32
#define PNUM  (WDIM*HDIM)   // 1024 spatial points
#define KCLU  64
#define PWID  10            // pooled map 10x10
#define PPTS  (PWID*PWID)   // 100
#define PTOT  (N_IMG*PPTS)  // 3200 batched pooled points = 200 x 16
#define FCH   84            // 32+32+20 concat channels
#define EPSN  1e-12f

typedef float v2f __attribute__((ext_vector_type(2)));
typedef float v8f __attribute__((ext_vector_type(8)));

// ---- gfx1250 async global->LDS copy (CDNA5 ISA) ----
// Per-lane: 16 bytes from global VADDR go to LDS at the VDST offset; tracked by ASYNCcnt.
// Low 32 bits of a generic LDS pointer are the in-allocation byte offset.
__device__ __forceinline__ void async_copy_b128(const float* g, float* l) {
    unsigned lds_off = (unsigned)(unsigned long long)l;
    asm volatile("global_load_async_to_lds_b128 %0, %1, off"
                 :: "v"(lds_off), "v"(g) : "memory");
}
__device__ __forceinline__ void wait_async0() {
    asm volatile("s_wait_asynccnt 0x0" ::: "memory");
}

// ---------------- 1) avg_pool2d(x, 3, stride 3, VALID) ----------------
__global__ void avgpool_k(const float* __restrict__ x, float* __restrict__ y) {
    int idx = blockIdx.x * blockDim.x + threadIdx.x;   // over N*C*100
    if (idx >= N_IMG * C_IN * PPTS) return;
    int j  = idx % PWID;
    int i  = (idx / PWID) % PWID;
    int nc = idx / PPTS;
    const float* xp = x + ((size_t)nc * WDIM + 3 * i) * HDIM + 3 * j;
    float s = 0.0f;
#pragma unroll
    for (int di = 0; di < 3; ++di)
#pragma unroll
        for (int dj = 0; dj < 3; ++dj)
            s += xp[di * HDIM + dj];
    y[idx] = s * (1.0f / 9.0f);
}

// ---------------- 2) SAME conv + ReLU as per-offset K-split WMMA GEMM ----------------
template <int KS, int OC>
__global__ __launch_bounds__(256) void conv_wmma_k(const float* __restrict__ y,
                                                   const float* __restrict__ w,
                                                   const float* __restrict__ b,
                                                   float* __restrict__ feat, int chOff) {
    constexpr int OCT = (OC + 15) / 16;
    constexpr int PAD = KS / 2;
    constexpr int KK  = KS * KS;
    int wave = blockIdx.x * (blockDim.x >> 5) + (threadIdx.x >> 5);
    int lane = threadIdx.x & 31;
    int ptile = wave % (PTOT / 16);
    int otile = wave / (PTOT / 16);
    if (otile >= OCT) return;                   // grids exact; wave-uniform
    int half = lane >> 4;
    int lid  = lane & 15;
    int col = ptile * 16 + lid;
    int n   = col / PPTS;
    int p   = col % PPTS;
    int pi  = p / PWID;
    int pj  = p % PWID;
    int row_a = otile * 16 + lid;
    int row_e = row_a < OC ? row_a : OC - 1;
    const float* Abase = w + ((size_t)row_e * C_IN + 2 * half) * KK;
    const float* Bbase = y + ((size_t)n * C_IN + 2 * half) * PPTS;
    v8f acc = {};
    for (int r = 0; r < KS; ++r) {
        int ii = pi + r - PAD;
        bool vi = (unsigned)ii < (unsigned)PWID;
        for (int s = 0; s < KS; ++s) {
            int jj = pj + s - PAD;
            bool valid = vi && ((unsigned)jj < (unsigned)PWID);
            int sp   = ii * PWID + jj;
            int woff = r * KS + s;
#pragma unroll 4
            for (int c = 0; c < C_IN; c += 4) {
                v2f a, bb;
                a.x  = Abase[(size_t)c * KK + woff];
                a.y  = Abase[(size_t)(c + 1) * KK + woff];
                bb.x = valid ? Bbase[(size_t)c * PPTS + sp] : 0.0f;
                bb.y = valid ? Bbase[(size_t)(c + 1) * PPTS + sp] : 0.0f;
                acc = __builtin_amdgcn_wmma_f32_16x16x4_f32(false, a, false, bb,
                                                            (short)0, acc, false, false);
            }
        }
    }
#pragma unroll
    for (int v = 0; v < 8; ++v) {
        int mrow = otile * 16 + v + 8 * half;
        if (mrow < OC)
            feat[((size_t)n * FCH + chOff + mrow) * PPTS + p] =
                fmaxf(acc[v] + b[mrow], 0.0f);
    }
}

// ---------------- 3) 1x1 "acc" conv: 84ch -> 1ch on 10x10 ----------------
__global__ void accconv_k(const float* __restrict__ feat, const float* __restrict__ aw,
                          const float* __restrict__ ab, float* __restrict__ accm) {
    int idx = blockIdx.x * blockDim.x + threadIdx.x;   // N*100
    if (idx >= N_IMG * PPTS) return;
    int p = idx % PPTS;
    int n = idx / PPTS;
    const float* f = feat + (size_t)n * FCH * PPTS + p;
    float s = ab[0];
#pragma unroll 4
    for (int ch = 0; ch < FCH; ++ch) s = fmaf(aw[ch], f[ch * PPTS], s);
    accm[idx] = s;
}

// ---------------- 4) bilinear upsample 10x10 -> 32x32, align_corners ----------------
__global__ void upsample_k(const float* __restrict__ accm, float* __restrict__ ctx) {
    int idx = blockIdx.x * blockDim.x + threadIdx.x;   // N*1024
    if (idx >= N_IMG * PNUM) return;
    int q  = idx % HDIM;
    int pq = (idx / HDIM) % WDIM;
    int n  = idx / PNUM;
    float cw = pq * (float)(PWID - 1) / (float)(WDIM - 1);
    float ch = q  * (float)(PWID - 1) / (float)(HDIM - 1);
    int i0 = min((int)floorf(cw), PWID - 1);
    int j0 = min((int)floorf(ch), PWID - 1);
    int i1 = min(i0 + 1, PWID - 1);
    int j1 = min(j0 + 1, PWID - 1);
    float wx = cw - (float)i0, wy = ch - (float)j0;
    const float* a = accm + (size_t)n * PPTS;
    ctx[idx] = (1.0f - wx) * ((1.0f - wy) * a[i0 * PWID + j0] + wy * a[i0 * PWID + j1])
             +          wx * ((1.0f - wy) * a[i1 * PWID + j0] + wy * a[i1 * PWID + j1]);
}

// ---------------- 5) channel-wise L2 normalize x -> xn ----------------
__global__ void l2norm_ch_k(const float* __restrict__ x, float* __restrict__ xn) {
    int idx = blockIdx.x * blockDim.x + threadIdx.x;   // N*1024
    if (idx >= N_IMG * PNUM) return;
    int p = idx % PNUM;
    int n = idx / PNUM;
    const float* xp = x + (size_t)n * C_IN * PNUM + p;
    float ss = 0.0f;
#pragma unroll 4
    for (int c = 0; c < C_IN; ++c) { float v = xp[c * PNUM]; ss = fmaf(v, v, ss); }
    float inv = 1.0f / fmaxf(sqrtf(ss), EPSN);
    float* op = xn + (size_t)n * C_IN * PNUM + p;
#pragma unroll 4
    for (int c = 0; c < C_IN; ++c) op[c * PNUM] = xp[c * PNUM] * inv;
}

// ---------------- 6) assign GEMM (WMMA f32, LDS-staged B) ----------------
// Block = 4 waves = the 4 K-tiles (full M=64); they share one 16x64 B chunk
// staged per step into LDS via async global->LDS copies. B L2 traffic /4.
__global__ __launch_bounds__(128) void assign_gemm_k(const float* __restrict__ xn,
                                                     const float* __restrict__ aw,
                                                     const float* __restrict__ ab,
                                                     float* __restrict__ sa) {
    __shared__ float Bs[16 * 64];                  // c-chunk(16) x cols(64)
    int ktile = threadIdx.x >> 5;                  // wave id = K-tile
    int lane  = threadIdx.x & 31;
    int pgrp  = blockIdx.x % (PNUM / 64);          // 16 col-groups
    int n     = blockIdx.x / (PNUM / 64);
    int half  = lane >> 4;
    int lid   = lane & 15;
    const float* Arow = aw + (size_t)(ktile * 16 + lid) * C_IN + half * 2;
    const float* Bsrc = xn + (size_t)n * C_IN * PNUM + pgrp * 64;
    v8f acc0 = {}, acc1 = {}, acc2 = {}, acc3 = {};
    for (int c0 = 0; c0 < C_IN; c0 += 16) {
        __syncthreads();                           // protect prior reads
        // stage 16x64 chunk: 256 float4, 2 per thread, coalesced along cols
#pragma unroll
        for (int q = 0; q < 2; ++q) {
            int e  = q * 128 + threadIdx.x;
            int cc = e >> 4, cg = e & 15;
            async_copy_b128(Bsrc + (size_t)(c0 + cc) * PNUM + cg * 4,
                            &Bs[cc * 64 + cg * 4]);
        }
        wait_async0();
        __syncthreads();
#pragma unroll
        for (int cc = 0; cc < 16; cc += 4) {
            v2f a;
            a.x = Arow[c0 + cc];
            a.y = Arow[c0 + cc + 1];
            const float* B0 = &Bs[(cc + 2 * half) * 64 + lid];
            v2f b0, b1, b2, b3;
            b0.x = B0[0];  b0.y = B0[64];
            b1.x = B0[16]; b1.y = B0[80];
            b2.x = B0[32]; b2.y = B0[96];
            b3.x = B0[48]; b3.y = B0[112];
            acc0 = __builtin_amdgcn_wmma_f32_16x16x4_f32(false, a, false, b0, (short)0, acc0, false, false);
            acc1 = __builtin_amdgcn_wmma_f32_16x16x4_f32(false, a, false, b1, (short)0, acc1, false, false);
            acc2 = __builtin_amdgcn_wmma_f32_16x16x4_f32(false, a, false, b2, (short)0, acc2, false, false);
            acc3 = __builtin_amdgcn_wmma_f32_16x16x4_f32(false, a, false, b3, (short)0, acc3, false, false);
        }
    }
    float* out = sa + ((size_t)n * KCLU + ktile * 16) * PNUM + pgrp * 64 + lid;
    v8f* accs[4] = {&acc0, &acc1, &acc2, &acc3};
#pragma unroll
    for (int t = 0; t < 4; ++t)
#pragma unroll
        for (int v = 0; v < 8; ++v) {
            int m = v + 8 * half;
            out[(size_t)m * PNUM + t * 16] = (*accs[t])[v] + ab[ktile * 16 + m];
        }
}

// ---------------- 7) softmax over K then multiply by context weight ----------------
__global__ void softmax_ctx_k(float* __restrict__ sa, const float* __restrict__ ctx) {
    int idx = blockIdx.x * blockDim.x + threadIdx.x;   // N*1024
    if (idx >= N_IMG * PNUM) return;
    int p = idx % PNUM;
    int n = idx / PNUM;
    float* s = sa + (size_t)n * KCLU * PNUM + p;
    float m = -3.0e38f;
    for (int k = 0; k < KCLU; ++k) m = fmaxf(m, s[(size_t)k * PNUM]);
    float sum = 0.0f;
    for (int k = 0; k < KCLU; ++k) sum += __expf(s[(size_t)k * PNUM] - m);
    float scale = ctx[idx] / sum;
    for (int k = 0; k < KCLU; ++k)
        s[(size_t)k * PNUM] = __expf(s[(size_t)k * PNUM] - m) * scale;
}

// ---------------- 8) per-(n,k) sum of reweighted assignments ----------------
__global__ void sasum_k(const float* __restrict__ sa, float* __restrict__ ssum) {
    int idx = blockIdx.x * blockDim.x + threadIdx.x;   // N*K
    if (idx >= N_IMG * KCLU) return;
    const float* s = sa + (size_t)idx * PNUM;
    float t = 0.0f;
#pragma unroll 4
    for (int p = 0; p < PNUM; ++p) t += s[p];
    ssum[idx] = t;
}

// ---------------- 9) VLAD GEMM (WMMA f32, LDS-staged transposed B) ----------------
// vlad[n,k,c] = sum_p sa[n,k,p] * xn[n,c,p]. Block = 4 K-tile waves sharing one
// 16(p) x 64(c) B chunk, staged TRANSPOSED ([c][p]) so fragment reads are
// contiguous pairs and global copies stay coalesced along p.
__global__ __launch_bounds__(128) void vlad_gemm_k(const float* __restrict__ sa,
                                                   const float* __restrict__ xn,
                                                   float* __restrict__ vlad) {
    __shared__ float Bs[64 * 16];                  // [ccol][pp]
    int ktile = threadIdx.x >> 5;
    int lane  = threadIdx.x & 31;
    int cgrp  = blockIdx.x % (C_IN / 64);          // 8 C-col groups
    int n     = blockIdx.x / (C_IN / 64);
    int half  = lane >> 4;
    int lid   = lane & 15;
    const float* Ap = sa + ((size_t)n * KCLU + ktile * 16 + lid) * PNUM + half * 2;
    const float* Bsrc = xn + ((size_t)n * C_IN + cgrp * 64) * PNUM;
    v8f acc0 = {}, acc1 = {}, acc2 = {}, acc3 = {};
    for (int p0 = 0; p0 < PNUM; p0 += 16) {
        __syncthreads();
        // stage 64 c-rows x 16 p's: 256 float4, 2 per thread (p-contiguous)
#pragma unroll
        for (int q = 0; q < 2; ++q) {
            int e    = q * 128 + threadIdx.x;
            int ccol = e >> 2, pp4 = e & 3;
            async_copy_b128(Bsrc + (size_t)ccol * PNUM + p0 + pp4 * 4,
                            &Bs[ccol * 16 + pp4 * 4]);
        }
        wait_async0();
        __syncthreads();
#pragma unroll
        for (int pp = 0; pp < 16; pp += 4) {
            v2f a;
            a.x = Ap[p0 + pp];
            a.y = Ap[p0 + pp + 1];
            int ppo = pp + 2 * half;
            v2f b0, b1, b2, b3;
            b0.x = Bs[(lid)      * 16 + ppo]; b0.y = Bs[(lid)      * 16 + ppo + 1];
            b1.x = Bs[(lid + 16) * 16 + ppo]; b1.y = Bs[(lid + 16) * 16 + ppo + 1];
            b2.x = Bs[(lid + 32) * 16 + ppo]; b2.y = Bs[(lid + 32) * 16 + ppo + 1];
            b3.x = Bs[(lid + 48) * 16 + ppo]; b3.y = Bs[(lid + 48) * 16 + ppo + 1];
            acc0 = __builtin_amdgcn_wmma_f32_16x16x4_f32(false, a, false, b0, (short)0, acc0, false, false);
            acc1 = __builtin_amdgcn_wmma_f32_16x16x4_f32(false, a, false, b1, (short)0, acc1, false, false);
            acc2 = __builtin_amdgcn_wmma_f32_16x16x4_f32(false, a, false, b2, (short)0, acc2, false, false);
            acc3 = __builtin_amdgcn_wmma_f32_16x16x4_f32(false, a, false, b3, (short)0, acc3, false, false);
        }
    }
    float* out = vlad + ((size_t)n * KCLU + ktile * 16) * C_IN + cgrp * 64 + lid;
    v8f* accs[4] = {&acc0, &acc1, &acc2, &acc3};
#pragma unroll
    for (int t = 0; t < 4; ++t)
#pragma unroll
        for (int v = 0; v < 8; ++v)
            out[(size_t)(v + 8 * half) * C_IN + t * 16] = (*accs[t])[v];
}

// ---------------- 10) centroid correction + intra-normalization per (n,k) ----------
__global__ void vlad_postnorm_k(float* __restrict__ vlad, const float* __restrict__ ssum,
                                const float* __restrict__ cent) {
    int idx = blockIdx.x * blockDim.x + threadIdx.x;   // N*K
    if (idx >= N_IMG * KCLU) return;
    int k = idx % KCLU;
    float s = ssum[idx];
    float* v = vlad + (size_t)idx * C_IN;
    const float* ck = cent + (size_t)k * C_IN;
    float ssq = 0.0f;
#pragma unroll 4
    for (int c = 0; c < C_IN; ++c) {
        float t = v[c] - s * ck[c];
        v[c] = t;
        ssq = fmaf(t, t, ssq);
    }
    float inv = 1.0f / fmaxf(sqrtf(ssq), EPSN);
#pragma unroll 4
    for (int c = 0; c < C_IN; ++c) v[c] *= inv;
}

// ---------------- 11) global L2 normalize per image -> d_out ----------------
__global__ void final_norm_k(const float* __restrict__ vlad, float* __restrict__ out) {
    int n = blockIdx.x;
    int t = threadIdx.x;
    __shared__ float red[256];
    const float* v = vlad + (size_t)n * KCLU * C_IN;
    float ss = 0.0f;
    for (int i = t; i < KCLU * C_IN; i += 256) { float a = v[i]; ss = fmaf(a, a, ss); }
    red[t] = ss;
    __syncthreads();
    for (int s = 128; s > 0; s >>= 1) {
        if (t < s) red[t] += red[t + s];
        __syncthreads();
    }
    float inv = 1.0f / fmaxf(sqrtf(red[0]), EPSN);
    float* o = out + (size_t)n * KCLU * C_IN;
    for (int i = t; i < KCLU * C_IN; i += 256) o[i] = v[i] * inv;
}

// ---------------- host-side orchestration ----------------
extern "C" void kernel_launch(void* const* d_in, const int* in_sizes, int n_in,
                              void* d_out, int out_size, void* d_ws, size_t ws_size,
                              hipStream_t stream) {
    const float* x       = (const float*)d_in[0];
    const float* conv1_w = (const float*)d_in[1];
    const float* conv1_b = (const float*)d_in[2];
    const float* conv2_w = (const float*)d_in[3];
    const float* conv2_b = (const float*)d_in[4];
    const float* conv3_w = (const float*)d_in[5];
    const float* conv3_b = (const float*)d_in[6];
    const float* acc_w   = (const float*)d_in[7];
    const float* acc_b   = (const float*)d_in[8];
    const float* assn_w  = (const float*)d_in[9];
    const float* assn_b  = (const float*)d_in[10];
    const float* cent    = (const float*)d_in[11];
    float* outp = (float*)d_out;

    // workspace layout (floats)
    float* ws = (float*)d_ws;
    size_t off = 0;
    float* y    = ws + off; off += (size_t)N_IMG * C_IN * PPTS;   // pooled 10x10
    float* feat = ws + off; off += (size_t)N_IMG * FCH * PPTS;    // 84-ch features
    float* accm = ws + off; off += (size_t)N_IMG * PPTS;          // context 10x10
    float* ctx  = ws + off; off += (size_t)N_IMG * PNUM;          // context 32x32
    float* xn   = ws + off; off += (size_t)N_IMG * C_IN * PNUM;   // normalized x
    float* sa   = ws + off; off += (size_t)N_IMG * KCLU * PNUM;   // soft assign
    float* ssum = ws + off; off += (size_t)N_IMG * KCLU;          // per-(n,k) sums
    float* vlad = ws + off; off += (size_t)N_IMG * KCLU * C_IN;   // vlad accum
    (void)ws_size; (void)n_in; (void)in_sizes; (void)out_size;

    const int T = 256;
    auto blocks = [](long long n, int t) { return (int)((n + t - 1) / t); };

    // 1) avg pool
    avgpool_k<<<blocks((long long)N_IMG * C_IN * PPTS, T), T, 0, stream>>>(x, y);
    // 2) convs + relu via WMMA (per-offset K-split GEMM)
    conv_wmma_k<3, 32><<<(2 * (PTOT / 16)) / 8, T, 0, stream>>>(y, conv1_w, conv1_b, feat, 0);
    conv_wmma_k<5, 32><<<(2 * (PTOT / 16)) / 8, T, 0, stream>>>(y, conv2_w, conv2_b, feat, 32);
    conv_wmma_k<7, 20><<<(2 * (PTOT / 16)) / 8, T, 0, stream>>>(y, conv3_w, conv3_b, feat, 64);
    // 3) 1x1 accumulate conv
    accconv_k<<<blocks((long long)N_IMG * PPTS, T), T, 0, stream>>>(feat, acc_w, acc_b, accm);
    // 4) bilinear upsample to 32x32
    upsample_k<<<blocks((long long)N_IMG * PNUM, T), T, 0, stream>>>(accm, ctx);
    // 5) channel L2 norm
    l2norm_ch_k<<<blocks((long long)N_IMG * PNUM, T), T, 0, stream>>>(x, xn);
    // 6) assign GEMM (WMMA + async LDS staging): block = 4 waves, grid = n*pgrp
    assign_gemm_k<<<N_IMG * (PNUM / 64), 128, 0, stream>>>(xn, assn_w, assn_b, sa);
    // 7) softmax over K, reweight by context
    softmax_ctx_k<<<blocks((long long)N_IMG * PNUM, T), T, 0, stream>>>(sa, ctx);
    // 8) per-(n,k) assignment sums
    sasum_k<<<blocks((long long)N_IMG * KCLU, T), T, 0, stream>>>(sa, ssum);
    // 9) vlad GEMM (WMMA + async LDS staging): block = 4 waves, grid = n*cgrp
    vlad_gemm_k<<<N_IMG * (C_IN / 64), 128, 0, stream>>>(sa, xn, vlad);
    // 10) centroid correction + intra-norm
    vlad_postnorm_k<<<blocks((long long)N_IMG * KCLU, T), T, 0, stream>>>(vlad, ssum, cent);
    // 11) global norm -> output
    final_norm_k<<<N_IMG, T, 0, stream>>>(vlad, outp);
}